// DRRAA_40982577938580
// MI455X (gfx1250) — compile-verified
//
#include <hip/hip_runtime.h>
#include <math.h>

#define Nn 16384
#define Kk 32
#define Dd 16
#define Ss 2048
#define Ee 500000
#define EPSf 1e-6f

typedef __attribute__((ext_vector_type(2))) float v2f;
typedef __attribute__((ext_vector_type(8))) float v8f;

__device__ __forceinline__ float sigmoidf(float x) { return 1.f / (1.f + expf(-x)); }

// ---------------- 1. column softmax of Zp (K x N), axis=0 ----------------
__global__ __launch_bounds__(256) void softmax_kernel(const float* __restrict__ Zp,
                                                      float* __restrict__ Z) {
    int n = blockIdx.x * 256 + threadIdx.x;   // n < Nn
    float v[Kk];
    float mx = -3.4e38f;
    #pragma unroll
    for (int k = 0; k < Kk; ++k) { v[k] = Zp[k * Nn + n]; mx = fmaxf(mx, v[k]); }
    float s = 0.f;
    #pragma unroll
    for (int k = 0; k < Kk; ++k) { v[k] = expf(v[k] - mx); s += v[k]; }
    float inv = 1.f / s;
    #pragma unroll
    for (int k = 0; k < Kk; ++k) Z[k * Nn + n] = v[k] * inv;
}

// ---------------- 2. s_k = sum_n Z[k,n] * sigmoid(Gate[n,k]) ----------------
__global__ __launch_bounds__(256) void colsum_kernel(const float* __restrict__ Z,
                                                     const float* __restrict__ Gate,
                                                     float* __restrict__ csum) {
    int k = blockIdx.x;                       // k < Kk
    float acc = 0.f;
    for (int n = threadIdx.x; n < Nn; n += 256)
        acc += Z[k * Nn + n] * sigmoidf(Gate[n * Kk + k]);
    __shared__ float red[256];
    red[threadIdx.x] = acc;
    __syncthreads();
    for (int s = 128; s > 0; s >>= 1) {
        if (threadIdx.x < s) red[threadIdx.x] += red[threadIdx.x + s];
        __syncthreads();
    }
    if (threadIdx.x == 0) csum[k] = red[0];
}

// ------- 3. M = Zs @ Cs (32x32, inner S), then AZC = A @ M (16x32) -------
__global__ __launch_bounds__(1024) void mgemm_kernel(const float* __restrict__ Z,
                                                     const float* __restrict__ Gate,
                                                     const float* __restrict__ A,
                                                     const int* __restrict__ sidx,
                                                     const float* __restrict__ csum,
                                                     float* __restrict__ azc) {
    __shared__ float M[Kk * Kk];
    int i = threadIdx.x >> 5;                 // 0..31  (row of Zs)
    int j = threadIdx.x & 31;                 // 0..31  (col of Cs)
    float inv = 1.f / csum[j];
    float acc = 0.f;
    for (int s = 0; s < Ss; ++s) {
        int idx = sidx[s];
        float g = sigmoidf(Gate[idx * Kk + j]);
        acc += Z[i * Nn + idx] * (Z[j * Nn + idx] * g);
    }
    M[threadIdx.x] = acc * inv;               // M[i][j]
    __syncthreads();
    if (threadIdx.x < Dd * Kk) {
        int d = threadIdx.x >> 5, jj = threadIdx.x & 31;
        float a = 0.f;
        #pragma unroll
        for (int ii = 0; ii < Kk; ++ii) a += A[d * Kk + ii] * M[ii * Kk + jj];
        azc[d * Kk + jj] = a;
    }
}

// ---------------- 4. X[n][d] = sum_k AZC[d][k] * Z[k][n] ----------------
__global__ __launch_bounds__(256) void x_kernel(const float* __restrict__ Z,
                                                const float* __restrict__ azc,
                                                float* __restrict__ X) {
    __shared__ float As[Dd * Kk];
    for (int t = threadIdx.x; t < Dd * Kk; t += 256) As[t] = azc[t];
    __syncthreads();
    int n = blockIdx.x * 256 + threadIdx.x;
    float acc[Dd];
    #pragma unroll
    for (int d = 0; d < Dd; ++d) acc[d] = 0.f;
    for (int k = 0; k < Kk; ++k) {
        float zv = Z[k * Nn + n];
        #pragma unroll
        for (int d = 0; d < Dd; ++d) acc[d] += As[d * Kk + k] * zv;
    }
    #pragma unroll
    for (int d = 0; d < Dd; ++d) X[(size_t)n * Dd + d] = acc[d];
}

// ------- 5. gather sampled rows: Y, norms, row-sums, sampled beta -------
__global__ __launch_bounds__(256) void gather_kernel(const float* __restrict__ X,
                                                     const float* __restrict__ beta,
                                                     const int* __restrict__ sidx,
                                                     float* __restrict__ Y,
                                                     float* __restrict__ nrm,
                                                     float* __restrict__ rs,
                                                     float* __restrict__ bsm) {
    int s = blockIdx.x * 256 + threadIdx.x;   // s < Ss
    int idx = sidx[s];
    float n2 = 0.f, r = 0.f;
    #pragma unroll
    for (int d = 0; d < Dd; ++d) {
        float v = X[(size_t)idx * Dd + d];
        Y[(size_t)s * Dd + d] = v;
        n2 += v * v;
        r += v;
    }
    nrm[s] = n2; rs[s] = r; bsm[s] = beta[idx];
}

// ------- 6. pairwise tile kernel: Gram via V_WMMA_F32_16X16X4_F32 -------
// one wave per 16x16 tile of the 2048x2048 matrix; 8 waves/block
__global__ __launch_bounds__(256) void pairwise_kernel(const float* __restrict__ Y,
                                                       const float* __restrict__ nrm,
                                                       const float* __restrict__ rs,
                                                       const float* __restrict__ bsm,
                                                       float* __restrict__ part) {
    const int lane = threadIdx.x & 31;
    const int gw   = blockIdx.x * 8 + (threadIdx.x >> 5);  // tile id 0..16383
    const int i0   = (gw >> 7) * 16;
    const int j0   = (gw & 127) * 16;
    const int m    = lane & 15;           // row (A) / col (B) index inside tile
    const int hi   = lane >> 4;           // lanes 16-31 hold K-offset +2
    const int kofs = 2 * hi;

    // Gram tile G = Yi (16x16) * Yj^T (16x16) as 4 chained K=4 WMMAs.
    // For a Gram matrix, A-tile and B-tile per-lane layouts coincide:
    // lane -> entity index m, regs hold two consecutive K values.
    v8f c = {};
    #pragma unroll
    for (int kc = 0; kc < 4; ++kc) {
        int kb = kc * 4 + kofs;
        v2f a = *(const v2f*)(Y + (size_t)(i0 + m) * Dd + kb);
        v2f b = *(const v2f*)(Y + (size_t)(j0 + m) * Dd + kb);
        c = __builtin_amdgcn_wmma_f32_16x16x4_f32(false, a, false, b,
                                                  (short)0, c, false, false);
    }

    // C/D layout: lane L, vgpr r -> row M = r + 8*(L/16), col N = L%16
    int gj = j0 + m;
    float nj = nrm[gj], rj = rs[gj], bj = bsm[gj];
    float acc = 0.f;
    #pragma unroll
    for (int r = 0; r < 8; ++r) {
        int gi = i0 + r + 8 * hi;
        float d2 = nrm[gi] + nj - 2.f * c[r]
                 + 2.f * EPSf * (rs[gi] - rj) + (float)Dd * EPSf * EPSf;
        d2 = fmaxf(d2, 0.f);
        float dist = sqrtf(d2);
        if (gi != gj) acc += expf(bsm[gi] + bj - dist);
    }

    __shared__ float red[256];
    red[threadIdx.x] = acc;
    __syncthreads();
    for (int s = 128; s > 0; s >>= 1) {
        if (threadIdx.x < s) red[threadIdx.x] += red[threadIdx.x + s];
        __syncthreads();
    }
    if (threadIdx.x == 0) part[blockIdx.x] = red[0];
}

// ---------------- 7. edge kernel over E sparse pairs ----------------
__global__ __launch_bounds__(256) void edge_kernel(const float* __restrict__ X,
                                                   const float* __restrict__ beta,
                                                   const int* __restrict__ si,
                                                   const int* __restrict__ sj,
                                                   float* __restrict__ part) {
    float acc = 0.f;
    for (int e = blockIdx.x * 256 + threadIdx.x; e < Ee; e += 512 * 256) {
        int i = si[e], j = sj[e];
        const float4* xi = (const float4*)(X + (size_t)i * Dd);
        const float4* xj = (const float4*)(X + (size_t)j * Dd);
        float s = 0.f;
        #pragma unroll
        for (int q = 0; q < 4; ++q) {
            float4 a = xi[q], b = xj[q];
            float d0 = a.x - b.x + EPSf, d1 = a.y - b.y + EPSf;
            float d2 = a.z - b.z + EPSf, d3 = a.w - b.w + EPSf;
            s += d0 * d0 + d1 * d1 + d2 * d2 + d3 * d3;
        }
        acc += beta[i] + beta[j] - sqrtf(s);
    }
    __shared__ float red[256];
    red[threadIdx.x] = acc;
    __syncthreads();
    for (int s = 128; s > 0; s >>= 1) {
        if (threadIdx.x < s) red[threadIdx.x] += red[threadIdx.x + s];
        __syncthreads();
    }
    if (threadIdx.x == 0) part[blockIdx.x] = red[0];
}

// ---------------- 8. final fixed-order reduce ----------------
__global__ __launch_bounds__(256) void final_kernel(const float* __restrict__ p1,
                                                    const float* __restrict__ p2,
                                                    float* __restrict__ out) {
    __shared__ float red[256];
    float a1 = 0.f, a2 = 0.f;
    for (int i = threadIdx.x; i < 2048; i += 256) a1 += p1[i];
    for (int i = threadIdx.x; i < 512;  i += 256) a2 += p2[i];

    red[threadIdx.x] = a1;
    __syncthreads();
    for (int s = 128; s > 0; s >>= 1) {
        if (threadIdx.x < s) red[threadIdx.x] += red[threadIdx.x + s];
        __syncthreads();
    }
    float tot1 = red[0];
    __syncthreads();

    red[threadIdx.x] = a2;
    __syncthreads();
    for (int s = 128; s > 0; s >>= 1) {
        if (threadIdx.x < s) red[threadIdx.x] += red[threadIdx.x + s];
        __syncthreads();
    }
    if (threadIdx.x == 0) {
        const float HALF_E2 = 3.6945280494653247f;  // 0.5 * e^2
        out[0] = red[0] - HALF_E2 * tot1;
    }
}

extern "C" void kernel_launch(void* const* d_in, const int* in_sizes, int n_in,
                              void* d_out, int out_size, void* d_ws, size_t ws_size,
                              hipStream_t stream) {
    const float* beta = (const float*)d_in[0];
    const float* A    = (const float*)d_in[1];
    const float* Zp   = (const float*)d_in[2];
    const float* Gate = (const float*)d_in[3];
    const int*   sidx = (const int*)d_in[4];
    const int*   si   = (const int*)d_in[5];
    const int*   sj   = (const int*)d_in[6];
    float* out = (float*)d_out;

    float* ws   = (float*)d_ws;                 // ~3.3 MB total, fits L2 easily
    float* Z    = ws;                           // K*N
    float* X    = Z + (size_t)Kk * Nn;          // N*D
    float* Y    = X + (size_t)Nn * Dd;          // S*D
    float* nrm  = Y + (size_t)Ss * Dd;          // S
    float* rs   = nrm + Ss;                     // S
    float* bsm  = rs + Ss;                      // S
    float* azc  = bsm + Ss;                     // D*K
    float* csum = azc + Dd * Kk;                // K
    float* p1   = csum + Kk;                    // 2048 partials
    float* p2   = p1 + 2048;                    // 512 partials

    softmax_kernel <<<Nn / 256, 256, 0, stream>>>(Zp, Z);
    colsum_kernel  <<<Kk,       256, 0, stream>>>(Z, Gate, csum);
    mgemm_kernel   <<<1,       1024, 0, stream>>>(Z, Gate, A, sidx, csum, azc);
    x_kernel       <<<Nn / 256, 256, 0, stream>>>(Z, azc, X);
    gather_kernel  <<<Ss / 256, 256, 0, stream>>>(X, beta, sidx, Y, nrm, rs, bsm);
    pairwise_kernel<<<2048,     256, 0, stream>>>(Y, nrm, rs, bsm, p1);
    edge_kernel    <<<512,      256, 0, stream>>>(X, beta, si, sj, p2);
    final_kernel   <<<1,        256, 0, stream>>>(p1, p2, out);
}